// GATv2Encoder_33861522162255
// MI455X (gfx1250) — compile-verified
//
#include <hip/hip_runtime.h>
#include <hip/hip_bf16.h>

typedef float v2f __attribute__((ext_vector_type(2)));
typedef float v8f __attribute__((ext_vector_type(8)));

#define FDIM 64
#define NEG_SLOPE 0.2f

// A tile LDS stride (floats): 68 -> bank(r,c) = (4r + c) % 64 : the 16 rows per
// lane half hit distinct banks, the two halves read different k-pairs (k%4 = 0,1 vs 2,3)
// -> conflict-free; 272 B row pitch keeps float4 stores 16B-aligned.
#define A_STRIDE 68

// ---------------------------------------------------------------------------
// Kernel 1: fused dual projection  xl = x @ wl,  xr = x @ wr
// x: [M, 64] row-major (M = G*N = 80000), w: [64, 64] row-major.
// One wave computes a 16(M) x 64(N) strip of BOTH outputs via
// v_wmma_f32_16x16x4_f32 (128 WMMAs / wave).
//
// W is stored in LDS in *fragment-native* layout: fragment (kk, nt) occupies
// 64 contiguous floats; lane L's v2f lives at frag*64 + L*2. A fragment read
// is then a single aligned ds_load_b64 into an even VGPR pair (no v_mov
// fix-ups, banks 2L/2L+1 -> conflict-free).
// ---------------------------------------------------------------------------
__global__ __launch_bounds__(256) void gat_proj_wmma(
    const float* __restrict__ x,
    const float* __restrict__ wl,
    const float* __restrict__ wr,
    float* __restrict__ xl,
    float* __restrict__ xr,
    int M)
{
    __shared__ __attribute__((aligned(16))) float s_wl[FDIM * FDIM];   // frag layout
    __shared__ __attribute__((aligned(16))) float s_wr[FDIM * FDIM];   // frag layout
    __shared__ __attribute__((aligned(16))) float s_a[8 * 16 * A_STRIDE];

    const int tid  = threadIdx.x;
    const int wave = tid >> 5;
    const int lane = tid & 31;

    // cooperative load of both weight matrices, swizzled to fragment layout:
    //   k = 4*kk + 2*half + comp,  n = 16*nt + (lane&15)
    //   addr = (kk*4 + nt)*64 + (half*16 + (n&15))*2 + comp
    for (int i = tid; i < FDIM * FDIM; i += 256) {
        const int k = i >> 6, n = i & 63;
        const int kk = k >> 2, kb = k & 3;
        const int comp = kb & 1, half = kb >> 1;
        const int addr = ((kk << 2) + (n >> 4)) * 64 + (half * 16 + (n & 15)) * 2 + comp;
        s_wl[addr] = wl[i];
        s_wr[addr] = wr[i];
    }

    const int row0 = (blockIdx.x * 8 + wave) * 16;
    const bool active = (row0 + 16) <= M;

    // per-wave A tile: 16 rows x 64 cols, coalesced float4 loads (32 f/lane)
    float* sa = s_a + wave * 16 * A_STRIDE;
    if (active) {
        const float4* xrow = reinterpret_cast<const float4*>(x + (size_t)row0 * FDIM);
        #pragma unroll
        for (int it = 0; it < 8; ++it) {
            int flat = it * 128 + lane * 4;       // 512B contiguous per iter
            float4 v = xrow[flat >> 2];
            int r = flat >> 6, c = flat & 63;     // c % 4 == 0 -> 16B aligned
            *reinterpret_cast<float4*>(sa + r * A_STRIDE + c) = v;
        }
    }
    __syncthreads();
    if (!active) return;

    const int lm = lane & 15;            // M index (A) / N index (B)
    const int kb = (lane >> 4) << 1;     // K sub-slab: lanes 0-15 -> 0, 16-31 -> 2

    const v2f* fwl = reinterpret_cast<const v2f*>(s_wl);
    const v2f* fwr = reinterpret_cast<const v2f*>(s_wr);

    v8f accL[4] = {};
    v8f accR[4] = {};

    #pragma unroll
    for (int kk = 0; kk < 16; ++kk) {    // K = 64 in steps of 4
        const int k0 = kk * 4 + kb;
        const v2f a = *reinterpret_cast<const v2f*>(sa + lm * A_STRIDE + k0);
        #pragma unroll
        for (int nt = 0; nt < 4; ++nt) { // N = 64 in 16-wide tiles
            const int fbase = ((kk << 2) + nt) * 32 + lane;  // v2f index
            const v2f bl = fwl[fbase];
            const v2f br = fwr[fbase];
            accL[nt] = __builtin_amdgcn_wmma_f32_16x16x4_f32(
                false, a, false, bl, (short)0, accL[nt], false, false);
            accR[nt] = __builtin_amdgcn_wmma_f32_16x16x4_f32(
                false, a, false, br, (short)0, accR[nt], false, false);
        }
    }

    // D layout: vgpr v, lanes 0-15 -> row v, lanes 16-31 -> row v+8; col = lane%16
    const int half = lane >> 4;
    #pragma unroll
    for (int nt = 0; nt < 4; ++nt) {
        #pragma unroll
        for (int v = 0; v < 8; ++v) {
            const size_t idx = (size_t)(row0 + v + half * 8) * FDIM + nt * 16 + lm;
            xl[idx] = accL[nt][v];
            xr[idx] = accR[nt][v];
        }
    }
}

// ---------------------------------------------------------------------------
// Kernel 2: init  (out = bias broadcast, max-keys = INT_MIN, denom = 0)
// ---------------------------------------------------------------------------
__global__ __launch_bounds__(256) void gat_init(
    int* __restrict__ mkey, float* __restrict__ denom,
    float* __restrict__ out, const float* __restrict__ bias,
    int NG, int outsz)
{
    int i = blockIdx.x * blockDim.x + threadIdx.x;
    if (i < outsz) out[i] = bias[i & 63];
    if (i < NG) { mkey[i] = (int)0x80000000; denom[i] = 0.0f; }
}

// monotone float <-> signed-int key for atomicMax on floats
__device__ __forceinline__ int f2key(float f) {
    int b = __float_as_int(f);
    return b >= 0 ? b : (b ^ 0x7fffffff);
}
__device__ __forceinline__ float key2f(int k) {
    int b = k >= 0 ? k : (k ^ 0x7fffffff);
    return __int_as_float(b);
}

__device__ __forceinline__ float lrelu(float v) {
    return v > 0.0f ? v : NEG_SLOPE * v;
}

// ---------------------------------------------------------------------------
// Kernel 3: edge logits + segment max.  One wave per (edge, graph) pair.
// ---------------------------------------------------------------------------
__global__ __launch_bounds__(256) void gat_logits(
    const float* __restrict__ xl, const float* __restrict__ xr,
    const long long* __restrict__ ei, const float* __restrict__ att,
    float* __restrict__ logit, int* __restrict__ mkey,
    int E, int ET, int Gg, int N)
{
    const int wid  = blockIdx.x * 8 + (threadIdx.x >> 5);
    const int lane = threadIdx.x & 31;
    if (wid >= ET * Gg) return;
    const int e = wid % ET;
    const int g = wid / ET;

    int s, d;
    if (e < E) { s = (int)ei[e]; d = (int)ei[E + e]; }
    else       { s = d = e - E; }          // self loop

    const float2* ps = reinterpret_cast<const float2*>(xl + ((size_t)g * N + s) * FDIM);
    const float2* pd = reinterpret_cast<const float2*>(xr + ((size_t)g * N + d) * FDIM);
    const float2  a  = reinterpret_cast<const float2*>(att)[lane];
    const float2  u  = ps[lane];
    const float2  v  = pd[lane];

    float r = lrelu(u.x + v.x) * a.x + lrelu(u.y + v.y) * a.y;
    #pragma unroll
    for (int o = 16; o > 0; o >>= 1) r += __shfl_xor(r, o, 32);

    if (lane == 0) {
        logit[wid] = r;
        atomicMax(&mkey[(size_t)d * Gg + g], f2key(r));
    }
}

// ---------------------------------------------------------------------------
// Kernel 4: ex = exp(logit - m[dst]); denom[dst] += ex.  One thread per pair.
// ---------------------------------------------------------------------------
__global__ __launch_bounds__(256) void gat_exp(
    const float* __restrict__ logit, float* __restrict__ exb,
    const int* __restrict__ mkey, float* __restrict__ denom,
    const long long* __restrict__ ei, int E, int ET, int Gg)
{
    int i = blockIdx.x * blockDim.x + threadIdx.x;
    if (i >= ET * Gg) return;
    const int e = i % ET;
    const int g = i / ET;
    const int d = (e < E) ? (int)ei[E + e] : (e - E);
    const float m  = key2f(mkey[(size_t)d * Gg + g]);
    const float ex = __expf(logit[i] - m);
    exb[i] = ex;
    atomicAdd(&denom[(size_t)d * Gg + g], ex);
}

// ---------------------------------------------------------------------------
// Kernel 5: out[g,dst,:] += (ex/denom[dst]) * xl[g,src,:].  Wave per pair.
// ---------------------------------------------------------------------------
__global__ __launch_bounds__(256) void gat_agg(
    const float* __restrict__ xl, const float* __restrict__ exb,
    const float* __restrict__ denom, const long long* __restrict__ ei,
    float* __restrict__ out, int E, int ET, int Gg, int N)
{
    const int wid  = blockIdx.x * 8 + (threadIdx.x >> 5);
    const int lane = threadIdx.x & 31;
    if (wid >= ET * Gg) return;
    const int e = wid % ET;
    const int g = wid / ET;

    int s, d;
    if (e < E) { s = (int)ei[e]; d = (int)ei[E + e]; }
    else       { s = d = e - E; }

    const float alpha = exb[wid] / denom[(size_t)d * Gg + g];
    const float2 v = reinterpret_cast<const float2*>(xl + ((size_t)g * N + s) * FDIM)[lane];
    float* po = out + ((size_t)g * N + d) * FDIM + lane * 2;
    atomicAdd(po,     alpha * v.x);
    atomicAdd(po + 1, alpha * v.y);
}

// ---------------------------------------------------------------------------
extern "C" void kernel_launch(void* const* d_in, const int* in_sizes, int n_in,
                              void* d_out, int out_size, void* d_ws, size_t ws_size,
                              hipStream_t stream) {
    const float*     x    = (const float*)d_in[0];     // [G, N, 64]
    const long long* ei   = (const long long*)d_in[1]; // [2, E] int64
    const float*     wl   = (const float*)d_in[2];     // [64, 64]
    const float*     wr   = (const float*)d_in[3];     // [64, 64]
    const float*     att  = (const float*)d_in[4];     // [1, 64]
    const float*     bias = (const float*)d_in[5];     // [64]
    float*           out  = (float*)d_out;             // [G, N, 64]

    const int N  = 1000;
    const int M  = in_sizes[0] / FDIM;   // 80000
    const int Gg = M / N;                // 80
    const int E  = in_sizes[1] / 2;      // 8000
    const int ET = E + N;                // 9000 (with self loops)
    const int pairs = ET * Gg;           // 720000

    // workspace carve-up
    char* p = (char*)d_ws;
    float* xl    = (float*)p;  p += (size_t)M * FDIM * sizeof(float);
    float* xr    = (float*)p;  p += (size_t)M * FDIM * sizeof(float);
    float* logit = (float*)p;  p += (size_t)pairs * sizeof(float);
    float* exb   = (float*)p;  p += (size_t)pairs * sizeof(float);
    int*   mkey  = (int*)p;    p += (size_t)N * Gg * sizeof(int);
    float* denom = (float*)p;

    // 1) dual projection GEMM (WMMA f32)
    const int projBlocks = (M / 16 + 7) / 8;
    gat_proj_wmma<<<projBlocks, 256, 0, stream>>>(x, wl, wr, xl, xr, M);

    // 2) init out/max/denom
    const int initN = out_size > N * Gg ? out_size : N * Gg;
    gat_init<<<(initN + 255) / 256, 256, 0, stream>>>(mkey, denom, out, bias,
                                                      N * Gg, out_size);

    // 3) logits + segment max   (wave per pair)
    gat_logits<<<(pairs + 7) / 8, 256, 0, stream>>>(xl, xr, ei, att, logit,
                                                    mkey, E, ET, Gg, N);

    // 4) exp + segment sum      (thread per pair)
    gat_exp<<<(pairs + 255) / 256, 256, 0, stream>>>(logit, exb, mkey, denom,
                                                     ei, E, ET, Gg);

    // 5) weighted scatter-add   (wave per pair)
    gat_agg<<<(pairs + 7) / 8, 256, 0, stream>>>(xl, exb, denom, ei, out,
                                                 E, ET, Gg, N);
}